// BahdanauAttention_70652212019656
// MI455X (gfx1250) — compile-verified
//
#include <hip/hip_runtime.h>
#include <stdint.h>

#define B_ 32
#define S_ 2048
#define H_ 1024
#define CH_ROWS 64                 // s-rows per chunk (64 * 4KB = 256KB LDS slab)
#define NCHUNK (S_ / CH_ROWS)      // 32 chunks per batch
#define TPAD 1028                  // LDS tile row stride in floats (pad 4 -> conflict-free WMMA A reads)

typedef __attribute__((ext_vector_type(2))) float v2f;
typedef __attribute__((ext_vector_type(4))) float v4f;
typedef __attribute__((ext_vector_type(8))) float v8f;
typedef __attribute__((ext_vector_type(4))) int   v4i;

typedef __attribute__((address_space(1))) v4i* gptr_v4i;   // global b128 payload
typedef __attribute__((address_space(3))) v4i* lptr_v4i;   // LDS b128 payload

__device__ __forceinline__ lptr_v4i to_lds_v4i(void* p) {
    // generic LDS pointer keeps the byte offset in the low 32 bits
    return (lptr_v4i)(uint32_t)(uintptr_t)p;
}
__device__ __forceinline__ gptr_v4i to_glob_v4i(const void* p) {
    return (gptr_v4i)(uintptr_t)p;
}

// ---------------------------------------------------------------------------
// K1: u[h] = sum_e W[e, H+h] * v[e]   (We^T @ v), 4 MB read, trivial.
// ---------------------------------------------------------------------------
__global__ void __launch_bounds__(256) k_make_u(const float* __restrict__ W,
                                                const float* __restrict__ v,
                                                float* __restrict__ u) {
    const int h = blockIdx.x * 256 + threadIdx.x;
    float acc = 0.0f;
#pragma unroll 8
    for (int e = 0; e < H_; ++e)
        acc += W[(size_t)e * (2 * H_) + H_ + h] * v[e];   // v[e] is block-uniform -> s_load
    u[h] = acc;
}

// ---------------------------------------------------------------------------
// K2: flash-style fused chunk kernel. One block = (batch b, 64-row s-chunk).
//  Phase 1: async-copy 64x1024 f32 slab of encoder_outputs to LDS (256KB).
//  Phase 2: energies e[s] = enc[s,:]·u via V_WMMA_F32_16X16X4_F32
//           (A = 16x4 enc tile, B = u[h..h+3] broadcast to all 16 columns,
//            so D[m, n] == e[s0+m] for every n; accumulate over K=H in C).
//  Phase 3: local max m_c and l_c = sum exp(e - m_c).
//  Phase 4: partial context p_c[h] = sum_s exp(e_s - m_c) * enc[s, h].
// ---------------------------------------------------------------------------
__global__ void __launch_bounds__(128) k_flash_chunk(const float* __restrict__ enc,
                                                     const float* __restrict__ u,
                                                     float* __restrict__ energies,
                                                     float* __restrict__ m_arr,
                                                     float* __restrict__ l_arr,
                                                     float* __restrict__ partials) {
    extern __shared__ float tile[];          // CH_ROWS * TPAD floats (dynamic, ~257KB)
    __shared__ float u_s[H_];
    __shared__ float e_s[CH_ROWS];
    __shared__ float p_s[CH_ROWS];

    const int t     = threadIdx.x;           // 0..127 (4 wave32)
    const int chunk = blockIdx.x;
    const int b     = blockIdx.y;
    const int s0    = chunk * CH_ROWS;
    const float* encb = enc + ((size_t)b * S_ + s0) * H_;

    // ---- Phase 1: stage slab (async to LDS) + u (direct) --------------------
#if __has_builtin(__builtin_amdgcn_global_load_async_to_lds_b128)
#pragma unroll 4
    for (int i = 0; i < (CH_ROWS * H_ / 4) / 128; ++i) {   // 128 b128 loads / thread
        const int idx = t + i * 128;
        const int row = idx >> 8;                          // 256 float4 per row
        const int c4  = idx & 255;
        __builtin_amdgcn_global_load_async_to_lds_b128(
            to_glob_v4i(encb + (size_t)row * H_ + c4 * 4),
            to_lds_v4i(&tile[row * TPAD + c4 * 4]),
            /*offset=*/0, /*cpol=*/0);
    }
#else
    for (int i = 0; i < (CH_ROWS * H_ / 4) / 128; ++i) {
        const int idx = t + i * 128;
        const int row = idx >> 8;
        const int c4  = idx & 255;
        *(v4f*)&tile[row * TPAD + c4 * 4] =
            *(const v4f*)(encb + (size_t)row * H_ + c4 * 4);
    }
#endif
#pragma unroll
    for (int i = 0; i < H_ / 128; ++i) u_s[t + i * 128] = u[t + i * 128];

#if __has_builtin(__builtin_amdgcn_global_load_async_to_lds_b128)
#if __has_builtin(__builtin_amdgcn_s_wait_asynccnt)
    __builtin_amdgcn_s_wait_asynccnt(0);
#else
    asm volatile("s_wait_asynccnt 0" ::: "memory");
#endif
#endif
    __syncthreads();

    // ---- Phase 2: energies via WMMA f32 16x16x4 ----------------------------
    const int lane = t & 31;
    const int wv   = t >> 5;                       // wave 0..3 -> rows 16wv..16wv+15
    const int mrow = wv * 16 + (lane & 15);        // A: lanes 0-15 hold M, K=0,1; lanes 16-31 K=2,3
    const int koff = (lane >> 4) * 2;

#if __has_builtin(__builtin_amdgcn_wmma_f32_16x16x4_f32)
    v8f acc = {0.f, 0.f, 0.f, 0.f, 0.f, 0.f, 0.f, 0.f};
#pragma unroll 4
    for (int h = 0; h < H_; h += 4) {
        v2f a  = *(const v2f*)&tile[mrow * TPAD + h + koff];  // A fragment (ds_load_b64)
        v2f bb = *(const v2f*)&u_s[h + koff];                 // B = u broadcast to all N
        acc = __builtin_amdgcn_wmma_f32_16x16x4_f32(false, a, false, bb,
                                                    (short)0, acc, false, false);
    }
    // D layout: VGPR j, lanes0-15 -> M=j ; lanes16-31 -> M=8+j. All N columns equal.
    if (lane == 0 || lane == 16) {
        const int rbase = wv * 16 + (lane >> 4) * 8;
#pragma unroll
        for (int j = 0; j < 8; ++j) {
            const float ev = acc[j];
            e_s[rbase + j] = ev;
            energies[(size_t)b * S_ + s0 + rbase + j] = ev;   // raw e, needed for weights
        }
    }
#else
    if (t < CH_ROWS) {
        float ev = 0.0f;
        for (int h = 0; h < H_; ++h) ev += tile[t * TPAD + h] * u_s[h];
        e_s[t] = ev;
        energies[(size_t)b * S_ + s0 + t] = ev;
    }
#endif
    __syncthreads();

    // ---- Phase 3: chunk-local softmax stats (all threads compute same) -----
    float mc = -3.402823466e38f;
#pragma unroll 8
    for (int s = 0; s < CH_ROWS; ++s) mc = fmaxf(mc, e_s[s]);
    float lc = 0.0f;
#pragma unroll 8
    for (int s = 0; s < CH_ROWS; ++s) lc += __expf(e_s[s] - mc);

    if (t == 0) {
        m_arr[b * NCHUNK + chunk] = mc;
        l_arr[b * NCHUNK + chunk] = lc;
    }
    if (t < CH_ROWS) p_s[t] = __expf(e_s[t] - mc);
    __syncthreads();

    // ---- Phase 4: partial context from LDS (8 h per thread) ----------------
    const int h0 = t * 8;
    v4f a0 = {0.f, 0.f, 0.f, 0.f}, a1 = {0.f, 0.f, 0.f, 0.f};
#pragma unroll 4
    for (int s = 0; s < CH_ROWS; ++s) {
        const float p = p_s[s];
        a0 += p * *(const v4f*)&tile[s * TPAD + h0];
        a1 += p * *(const v4f*)&tile[s * TPAD + h0 + 4];
    }
    float* pp = partials + (size_t)(b * NCHUNK + chunk) * H_ + h0;
    *(v4f*)pp       = a0;
    *(v4f*)(pp + 4) = a1;
}

// ---------------------------------------------------------------------------
// K3: merge 32 chunks per batch (fixed-order = deterministic) and emit
//     context (d_out[0 .. B*H)) and attn_weights (d_out[B*H .. B*H + B*S)).
// ---------------------------------------------------------------------------
__global__ void __launch_bounds__(256) k_finalize(const float* __restrict__ m_arr,
                                                  const float* __restrict__ l_arr,
                                                  const float* __restrict__ partials,
                                                  const float* __restrict__ energies,
                                                  float* __restrict__ out) {
    const int b = blockIdx.x;
    const int t = threadIdx.x;

    float M = -3.402823466e38f;
#pragma unroll
    for (int c = 0; c < NCHUNK; ++c) M = fmaxf(M, m_arr[b * NCHUNK + c]);
    float L = 0.0f;
#pragma unroll
    for (int c = 0; c < NCHUNK; ++c)
        L += __expf(m_arr[b * NCHUNK + c] - M) * l_arr[b * NCHUNK + c];
    const float invL = 1.0f / L;

    // context[b, h]  (256 threads x 4 h)
    const int h0 = t * 4;
    v4f acc = {0.f, 0.f, 0.f, 0.f};
#pragma unroll
    for (int c = 0; c < NCHUNK; ++c) {
        const float sc = __expf(m_arr[b * NCHUNK + c] - M);
        acc += sc * *(const v4f*)&partials[(size_t)(b * NCHUNK + c) * H_ + h0];
    }
    acc *= invL;
    *(v4f*)&out[(size_t)b * H_ + h0] = acc;

    // attn_weights[b, s]
    for (int s = t; s < S_; s += 256)
        out[(size_t)B_ * H_ + (size_t)b * S_ + s] =
            __expf(energies[(size_t)b * S_ + s] - M) * invL;
}

// ---------------------------------------------------------------------------
extern "C" void kernel_launch(void* const* d_in, const int* in_sizes, int n_in,
                              void* d_out, int out_size, void* d_ws, size_t ws_size,
                              hipStream_t stream) {
    // inputs: 0 hidden_state (unused: softmax shift-invariance), 1 encoder_outputs,
    //         2 W, 3 b (unused: shift-invariance), 4 v, 5 batch, 6 seqlen
    const float* enc = (const float*)d_in[1];
    const float* W   = (const float*)d_in[2];
    const float* v   = (const float*)d_in[4];

    float* ws       = (float*)d_ws;
    float* u        = ws;                               // 1024
    float* energies = u + H_;                           // 65536
    float* m_arr    = energies + (size_t)B_ * S_;       // 1024
    float* l_arr    = m_arr + B_ * NCHUNK;              // 1024
    float* partials = l_arr + B_ * NCHUNK;              // 32*32*1024 = 1M floats

    k_make_u<<<dim3(H_ / 256), dim3(256), 0, stream>>>(W, v, u);

    const size_t ldsBytes = (size_t)CH_ROWS * TPAD * sizeof(float);  // ~257 KB (CDNA5: 320KB/WGP)
    k_flash_chunk<<<dim3(NCHUNK, B_), dim3(128), ldsBytes, stream>>>(
        enc, u, energies, m_arr, l_arr, partials);

    k_finalize<<<dim3(B_), dim3(256), 0, stream>>>(m_arr, l_arr, partials, energies,
                                                   (float*)d_out);
}